// FPSsubsample_56315611185334
// MI455X (gfx1250) — compile-verified
//
#include <hip/hip_runtime.h>

#define BS 4
#define N  4096
#define M  1024
#define C  128

typedef unsigned int u32x4 __attribute__((ext_vector_type(4)));
typedef int          i32x8 __attribute__((ext_vector_type(8)));
typedef int          i32x4 __attribute__((ext_vector_type(4)));

#if defined(__has_builtin)
#  if __has_builtin(__builtin_amdgcn_tensor_load_to_lds) && __has_builtin(__builtin_amdgcn_s_wait_tensorcnt)
#    define USE_TDM 1
#  endif
#endif
#ifndef USE_TDM
#  define USE_TDM 0
#endif

// ---------------------------------------------------------------------------
// threefry2x32 (JAX PRNG), key = jax.random.key(1) -> (hi=0, lo=1)
// ---------------------------------------------------------------------------
__device__ __forceinline__ unsigned rotl32(unsigned x, unsigned r) {
  return (x << r) | (x >> (32u - r));
}

__device__ void threefry2x32(unsigned ks0, unsigned ks1, unsigned x0, unsigned x1,
                             unsigned* o0, unsigned* o1) {
  const unsigned ks2 = ks0 ^ ks1 ^ 0x1BD11BDAu;
  const unsigned rotA[4] = {13u, 15u, 26u, 6u};
  const unsigned rotB[4] = {17u, 29u, 16u, 24u};
  x0 += ks0; x1 += ks1;
  #pragma unroll
  for (int i = 0; i < 4; ++i) { x0 += x1; x1 = rotl32(x1, rotA[i]); x1 ^= x0; }
  x0 += ks1; x1 += ks2 + 1u;
  #pragma unroll
  for (int i = 0; i < 4; ++i) { x0 += x1; x1 = rotl32(x1, rotB[i]); x1 ^= x0; }
  x0 += ks2; x1 += ks0 + 2u;
  #pragma unroll
  for (int i = 0; i < 4; ++i) { x0 += x1; x1 = rotl32(x1, rotA[i]); x1 ^= x0; }
  x0 += ks0; x1 += ks1 + 3u;
  #pragma unroll
  for (int i = 0; i < 4; ++i) { x0 += x1; x1 = rotl32(x1, rotB[i]); x1 ^= x0; }
  x0 += ks1; x1 += ks2 + 4u;
  #pragma unroll
  for (int i = 0; i < 4; ++i) { x0 += x1; x1 = rotl32(x1, rotA[i]); x1 ^= x0; }
  x0 += ks2; x1 += ks0 + 5u;
  *o0 = x0; *o1 = x1;
}

// ---------------------------------------------------------------------------
// Seed: farthest0[b] per batch. randint(key(1),(4,),0,4096): span=4096=2^12,
// multiplier 2^32%4096==0 -> a[b] = lower_bits[b] % 4096 = x1-output word b.
// Then k = a % msum; farthest0 = index of (k+1)-th set mask bit.
// ---------------------------------------------------------------------------
__global__ void fps_seed_kernel(const unsigned char* __restrict__ mask,
                                int* __restrict__ ws_far) {
  const int b = threadIdx.x;
  if (b >= BS) return;
  unsigned o0, o1;
  threefry2x32(0u, 1u, (unsigned)b, (unsigned)(b + 4), &o0, &o1);
  const unsigned a = o1 & (unsigned)(N - 1);
  const unsigned char* mb = mask + (size_t)b * N;
  int msum = 0;
  for (int i = 0; i < N; ++i) msum += (mb[i] != 0);
  if (msum <= 0) msum = 1;
  const unsigned k = a % (unsigned)msum;
  int c = 0, far = 0;
  for (int i = 0; i < N; ++i) {
    if (mb[i]) { if ((unsigned)c == k) { far = i; break; } ++c; }
  }
  ws_far[b] = far;
}

// ---------------------------------------------------------------------------
// FPS iterations. One block per batch, 1024 threads (32 wave32 waves).
// Per iteration: TDM DMA of row abq[b,far,:,:] (64KB) into LDS, distance
// update in registers, first-index argmax via shuffles + LDS, fixed-point
// early exit (min-update is idempotent once farthest repeats).
// ---------------------------------------------------------------------------
__global__ __launch_bounds__(1024) void fps_iter_kernel(
    const float* __restrict__ abq, const unsigned char* __restrict__ mask,
    const int* __restrict__ ws_far, int* __restrict__ qidx) {
  extern __shared__ float smem[];
  float* rowbuf = smem;                     // 16384 floats (64 KB)
  float* s_rv   = smem + 16384;             // 32 floats
  int*   s_ri   = (int*)(smem + 16416);     // 32 ints
  int*   s_far  = (int*)(smem + 16448);     // 1 int

  const int b    = blockIdx.x;
  const int t    = threadIdx.x;
  const int lane = t & 31;
  const int wid  = t >> 5;

  float dist[4];
  bool  mk[4];
  #pragma unroll
  for (int u = 0; u < 4; ++u) {
    dist[u] = 1e8f;
    mk[u]   = mask[(size_t)b * N + (t + 1024 * u)] != 0;
  }

  if (t == 0) *s_far = ws_far[b];
  __syncthreads();
  int far = *s_far;

#if USE_TDM
  const unsigned lds_off  = (unsigned)(uintptr_t)(void*)rowbuf;
  const uint64_t abq_base = (uint64_t)(uintptr_t)abq;
#endif

  for (int it = 0; it < M; ++it) {
    if (t == 0) qidx[b * M + it] = far;

#if USE_TDM
    if (t < 32) {  // wave 0 issues the DMA (EXEC ignored by TDM)
      const uint64_t ga =
          abq_base + (((uint64_t)b * N + (uint64_t)far) * (uint64_t)N) * 16ull;
      u32x4 g0;
      g0[0] = 1u;                                        // count=1, user mode
      g0[1] = lds_off;                                   // lds_addr (bytes)
      g0[2] = (unsigned)(ga & 0xffffffffull);            // global_addr[31:0]
      g0[3] = ((unsigned)(ga >> 32) & 0x01ffffffu) | 0x80000000u; // [56:32] | type=2
      i32x8 g1;
      g1[0] = (int)(2u << 16);                           // data_size = 4 bytes
      g1[1] = (int)((16384u & 0xffffu) << 16);           // tensor_dim0 lo16
      g1[2] = (int)((16384u >> 16) | (1u << 16));        // tensor_dim0 hi16 | tensor_dim1 lo16=1
      g1[3] = (int)(16384u << 16);                       // tensor_dim1 hi16=0 | tile_dim0=16384
      g1[4] = 1;                                         // tile_dim1=1, tile_dim2=0
      g1[5] = 16384;                                     // tensor_dim0_stride lo32
      g1[6] = (int)((16384u & 0xffffu) << 16);           // stride0 hi16=0 | stride1 lo16
      g1[7] = (int)(16384u >> 16);                       // stride1 hi
      const i32x4 gz4 = {0, 0, 0, 0};                    // groups 2/3 unused (<=2D)
      const i32x8 gz8 = {0, 0, 0, 0, 0, 0, 0, 0};       // extra group unused
      __builtin_amdgcn_tensor_load_to_lds(g0, g1, gz4, gz4, gz8, 0);
      __builtin_amdgcn_s_wait_tensorcnt(0);
    }
    asm volatile("" ::: "memory");
    __syncthreads();
#endif

    float bv = -3.402823466e38f;
    int   bi = 0x7fffffff;
    #pragma unroll
    for (int u = 0; u < 4; ++u) {
      const int j = t + 1024 * u;
#if USE_TDM
      const float4 v = ((const float4*)rowbuf)[j];
#else
      const float4 v = ((const float4*)abq)[((size_t)b * N + far) * N + j];
#endif
      float d = sqrtf(v.x * v.x + v.y * v.y + v.z * v.z + v.w * v.w);
      d = mk[u] ? -100.0f : d;              // faithful: where(mask, -100, d)
      const float nd = fminf(dist[u], d);
      dist[u] = nd;
      if (nd > bv || (nd == bv && j < bi)) { bv = nd; bi = j; }
    }

    // wave32 argmax, first-index tie-break
    #pragma unroll
    for (int off = 16; off > 0; off >>= 1) {
      const float ov = __shfl_down(bv, off, 32);
      const int   oi = __shfl_down(bi, off, 32);
      if (ov > bv || (ov == bv && oi < bi)) { bv = ov; bi = oi; }
    }
    if (lane == 0) { s_rv[wid] = bv; s_ri[wid] = bi; }
    __syncthreads();
    if (wid == 0) {
      bv = s_rv[lane]; bi = s_ri[lane];
      #pragma unroll
      for (int off = 16; off > 0; off >>= 1) {
        const float ov = __shfl_down(bv, off, 32);
        const int   oi = __shfl_down(bi, off, 32);
        if (ov > bv || (ov == bv && oi < bi)) { bv = ov; bi = oi; }
      }
      if (lane == 0) *s_far = bi;
    }
    __syncthreads();
    const int far_next = *s_far;
    if (far_next == far) {                   // fixed point: repeats forever
      for (int x = it + 1 + t; x < M; x += 1024) qidx[b * M + x] = far_next;
      break;
    }
    far = far_next;
  }
}

// ---------------------------------------------------------------------------
// sub_abq[b,i,j,:] = abq[b, q[i], q[j], :]   (float4 per element)
// one block per (b,i); q[] cached in LDS
// ---------------------------------------------------------------------------
__global__ __launch_bounds__(256) void gather_abq_kernel(
    const float* __restrict__ abq, const int* __restrict__ qidx,
    float* __restrict__ out_abq) {
  __shared__ int sq[M];
  const int bi = blockIdx.x;           // 0..BS*M-1
  const int b  = bi >> 10;
  const int i  = bi & (M - 1);
  const int t  = threadIdx.x;
  const int* q = qidx + b * M;
  #pragma unroll
  for (int u = 0; u < 4; ++u) sq[t + 256 * u] = q[t + 256 * u];
  __syncthreads();
  const int qi = sq[i];
  const float4* src = (const float4*)abq + ((size_t)b * N + qi) * N;
  float4*       dst = (float4*)out_abq + ((size_t)b * M + i) * M;
  #pragma unroll
  for (int u = 0; u < 4; ++u) {
    const int j = t + 256 * u;
    dst[j] = src[sq[j]];
  }
}

// ---------------------------------------------------------------------------
// sub_vals[b,i,:] = vals[b,q[i],:] ; sub_mask[b,i] = mask[b,q[i]]
// 32 lanes per row (128 f32 = 32 float4, contiguous)
// ---------------------------------------------------------------------------
__global__ __launch_bounds__(256) void gather_vals_mask_kernel(
    const float* __restrict__ vals, const unsigned char* __restrict__ mask,
    const int* __restrict__ qidx, float* __restrict__ out_vals,
    float* __restrict__ out_mask) {
  const int g    = blockIdx.x * 256 + threadIdx.x;  // BS*M*32 total
  const int r    = g >> 5;                          // row = b*M+i
  const int lane = g & 31;
  const int b    = r >> 10;
  const int q    = qidx[r];
  const float4* src = (const float4*)vals + ((size_t)b * N + q) * (C / 4);
  float4*       dst = (float4*)out_vals + (size_t)r * (C / 4);
  dst[lane] = src[lane];
  if (lane == 0) out_mask[r] = mask[(size_t)b * N + q] ? 1.0f : 0.0f;
}

// ---------------------------------------------------------------------------
extern "C" void kernel_launch(void* const* d_in, const int* in_sizes, int n_in,
                              void* d_out, int out_size, void* d_ws, size_t ws_size,
                              hipStream_t stream) {
  (void)in_sizes; (void)n_in; (void)out_size; (void)ws_size;
  const float*         abq  = (const float*)d_in[0];
  const float*         vals = (const float*)d_in[1];
  const unsigned char* mask = (const unsigned char*)d_in[2];

  float* out      = (float*)d_out;
  float* out_abq  = out;                                   // 4*1024*1024*4
  float* out_vals = out + (size_t)BS * M * M * 4;          // 4*1024*128
  float* out_mask = out_vals + (size_t)BS * M * C;         // 4*1024

  int* ws_far = (int*)d_ws;        // BS ints (padded to 8)
  int* qidx   = ws_far + 8;        // BS*M ints

  fps_seed_kernel<<<1, 32, 0, stream>>>(mask, ws_far);

  const size_t smem_bytes = (16384 + 32 + 32 + 4) * sizeof(float);
  fps_iter_kernel<<<BS, 1024, smem_bytes, stream>>>(abq, mask, ws_far, qidx);

  gather_abq_kernel<<<BS * M, 256, 0, stream>>>(abq, qidx, out_abq);

  gather_vals_mask_kernel<<<(BS * M * 32) / 256, 256, 0, stream>>>(
      vals, mask, qidx, out_vals, out_mask);
}